// GNN_2_7275674599612
// MI455X (gfx1250) — compile-verified
//
#include <hip/hip_runtime.h>

// ---------------------------------------------------------------------------
// 2-layer GCN forward for MI455X (gfx1250, wave32).
//   deg -> dinv -> xw = x@W1 (WMMA f32 16x16x4) -> h = scatter(norm*xw)+b1
//   -> relu -> hw = h@W2 (WMMA) -> out = scatter(norm*hw)+b2
// Scatter phase is L2-resident (xw/h/out all << 192MB L2). The scatter adds
// use inline-asm GLOBAL_ATOMIC_ADD_F32 (no-return, native fp32 L2 atomic) to
// guarantee we never fall back to a CAS retry loop.
// ---------------------------------------------------------------------------

typedef float v2f __attribute__((ext_vector_type(2)));
typedef float v8f __attribute__((ext_vector_type(8)));

#define GCN_N 100000
#define GCN_E 1600000

// ---------------- degree / norm ----------------
__global__ void fill_deg_k(int* __restrict__ deg, int n) {
    int i = blockIdx.x * blockDim.x + threadIdx.x;
    if (i < n) deg[i] = 1;  // self-loop contributes 1
}

__global__ void deg_scatter_k(const int* __restrict__ dst, int* __restrict__ deg, int e) {
    int i = blockIdx.x * blockDim.x + threadIdx.x;
    if (i < e) atomicAdd(&deg[dst[i]], 1);
}

__global__ void dinv_k(const int* __restrict__ deg, float* __restrict__ dinv, int n) {
    int i = blockIdx.x * blockDim.x + threadIdx.x;
    if (i < n) dinv[i] = rsqrtf((float)deg[i]);  // deg >= 1 always
}

// ---------------- GEMM: Y[N x COUT] = X[N x 128] @ W[128 x COUT] ----------------
// One wave per 16x16 output tile; K-loop of 32 x V_WMMA_F32_16X16X4_F32.
// A (16x4 f32): lanes 0-15 hold M=lane, K={0,1}; lanes 16-31 hold M=lane-16, K={2,3}.
// B (4x16 f32): N striped across lanes; same K split across lane halves / VGPRs.
// C/D (16x16 f32): vgpr i, lane L -> element ( (L>>4)*8 + i, L&15 ).
template <int COUT>
__global__ void gemm_wmma_k(const float* __restrict__ X, const float* __restrict__ W,
                            float* __restrict__ Y) {
    const int lane = threadIdx.x & 31;
    const int wave = threadIdx.x >> 5;            // 4 waves / block
    const int col0 = blockIdx.y * 64 + wave * 16; // 16-col tile per wave
    const int row0 = blockIdx.x * 16;             // 16-row tile (N % 16 == 0)

    const int mn = lane & 15;            // M (for A) / N (for B,D) within tile
    const int kb = (lane >> 4) * 2;      // K sub-offset for this lane half

    const float* __restrict__ xrow = X + (size_t)(row0 + mn) * 128;

    v8f c = {};
#pragma unroll 4
    for (int k = 0; k < 128; k += 4) {
        v2f a, b;
        a.x = xrow[k + kb];
        a.y = xrow[k + kb + 1];
        b.x = W[(size_t)(k + kb) * COUT + col0 + mn];
        b.y = W[(size_t)(k + kb + 1) * COUT + col0 + mn];
        // 8 args: (neg_a, A, neg_b, B, c_mod, C, reuse_a, reuse_b)
        c = __builtin_amdgcn_wmma_f32_16x16x4_f32(false, a, false, b, (short)0, c,
                                                  false, false);
    }

    const int mbase = (lane >> 4) * 8;
#pragma unroll
    for (int i = 0; i < 8; ++i) {
        Y[(size_t)(row0 + mbase + i) * COUT + col0 + mn] = c[i];
    }
}

// ---------------- bias broadcast init ----------------
template <int C>
__global__ void fill_bias_k(float* __restrict__ out, const float* __restrict__ bias,
                            long nelem) {
    long i = (long)blockIdx.x * blockDim.x + threadIdx.x;
    if (i < nelem) out[i] = bias[i & (C - 1)];
}

// Native fp32 global atomic add, no return (STOREcnt path, L2 RMW unit).
__device__ __forceinline__ void atomic_add_f32_native(float* p, float v) {
    asm volatile("global_atomic_add_f32 %0, %1, off" : : "v"(p), "v"(v) : "memory");
}

// ---------------- edge scatter-add: out[dst] += norm * feat[src] ----------------
// One wave32 per (edge or self-loop) row; VEC = C/32 floats per lane.
template <int C>
__global__ void scatter_add_k(const float* __restrict__ feat,
                              const int* __restrict__ src, const int* __restrict__ dst,
                              const float* __restrict__ dinv, float* __restrict__ out,
                              int nEdges, int nNodes) {
    constexpr int VEC = C / 32;
    const int lane = threadIdx.x & 31;
    const long idx = ((long)blockIdx.x * blockDim.x + threadIdx.x) >> 5;
    const long total = (long)nEdges + nNodes;
    if (idx >= total) return;

    int s, d;
    if (idx < nEdges) { s = src[idx]; d = dst[idx]; }
    else              { s = d = (int)(idx - nEdges); }   // self-loop

    const float w = dinv[s] * dinv[d];
    const float* __restrict__ fp = feat + (size_t)s * C + lane * VEC;
    float* __restrict__ op = out + (size_t)d * C + lane * VEC;
#pragma unroll
    for (int j = 0; j < VEC; ++j) {
        atomic_add_f32_native(op + j, fp[j] * w);
    }
}

// ---------------- relu ----------------
__global__ void relu_k(float* __restrict__ h, long n) {
    long i = (long)blockIdx.x * blockDim.x + threadIdx.x;
    if (i < n) h[i] = fmaxf(h[i], 0.0f);
}

// ---------------------------------------------------------------------------
extern "C" void kernel_launch(void* const* d_in, const int* in_sizes, int n_in,
                              void* d_out, int out_size, void* d_ws, size_t ws_size,
                              hipStream_t stream) {
    const float* x  = (const float*)d_in[0];
    const int*   ei = (const int*)d_in[1];   // [2][E] int32
    const float* W1 = (const float*)d_in[2];
    const float* b1 = (const float*)d_in[3];
    const float* W2 = (const float*)d_in[4];
    const float* b2 = (const float*)d_in[5];
    float* out = (float*)d_out;

    const int N = GCN_N;
    const int E = in_sizes[1] / 2;
    const int* src = ei;
    const int* dst = ei + E;

    // workspace carve-up (~103 MB): deg | dinv | xw | h ; hw reuses xw region
    char* ws = (char*)d_ws;
    size_t o = 0;
    auto align256 = [](size_t v) { return (v + 255) & ~(size_t)255; };
    int*   deg  = (int*)(ws + o);   o = align256(o + (size_t)N * 4);
    float* dinv = (float*)(ws + o); o = align256(o + (size_t)N * 4);
    float* xw   = (float*)(ws + o); o = align256(o + (size_t)N * 128 * 4);
    float* h    = (float*)(ws + o); o = align256(o + (size_t)N * 128 * 4);
    float* hw   = xw;  // safe: xw is dead after layer-1 scatter (stream-ordered)

    const int T = 256;
    const long rowsAll = (long)E + N;            // edges + self-loops
    const int scatterBlocks = (int)((rowsAll * 32 + T - 1) / T);

    // 1) degrees + symmetric norm
    fill_deg_k<<<(N + T - 1) / T, T, 0, stream>>>(deg, N);
    deg_scatter_k<<<(E + T - 1) / T, T, 0, stream>>>(dst, deg, E);
    dinv_k<<<(N + T - 1) / T, T, 0, stream>>>(deg, dinv, N);

    // 2) layer 1: xw = x @ W1  (6250 row tiles x 8 col tiles; 4 col tiles/block)
    gemm_wmma_k<128><<<dim3(N / 16, 2), 128, 0, stream>>>(x, W1, xw);

    // 3) h = b1; h[dst] += norm * xw[src]; relu
    fill_bias_k<128><<<(int)(((long)N * 128 + T - 1) / T), T, 0, stream>>>(h, b1,
                                                                (long)N * 128);
    scatter_add_k<128><<<scatterBlocks, T, 0, stream>>>(xw, src, dst, dinv, h, E, N);
    relu_k<<<(int)(((long)N * 128 + T - 1) / T), T, 0, stream>>>(h, (long)N * 128);

    // 4) layer 2: hw = h @ W2
    gemm_wmma_k<64><<<dim3(N / 16, 1), 128, 0, stream>>>(h, W2, hw);

    // 5) out = b2; out[dst] += norm * hw[src]
    fill_bias_k<64><<<(int)(((long)N * 64 + T - 1) / T), T, 0, stream>>>(out, b2,
                                                                (long)N * 64);
    scatter_add_k<64><<<scatterBlocks, T, 0, stream>>>(hw, src, dst, dinv, out, E, N);
}